// DifferentiableSparseHypergraph_59605556134253
// MI455X (gfx1250) — compile-verified
//
#include <hip/hip_runtime.h>
#include <math.h>

typedef float v2f __attribute__((ext_vector_type(2)));
typedef float v8f __attribute__((ext_vector_type(8)));

#define N_     64
#define C_     256
#define T_     64
#define V_     64
#define INTER_ 32
#define M_     128
#define K_TOP  10

// ---------------------------------------------------------------------------
// Kernel 1: x_mean[n,c,v] = mean_t x[n,c,t,v]
// One block per (n,c): reduces a contiguous 16KB tile with float4 (b128) loads.
// Pure streaming: this is the bandwidth-bound part (256 MB total).
// ---------------------------------------------------------------------------
__global__ __launch_bounds__(256) void mean_t_kernel(const float* __restrict__ x,
                                                     float* __restrict__ xm) {
  __shared__ float4 red[256];
  const int tid = threadIdx.x;
  const float4* xp4 = (const float4*)(x + (size_t)blockIdx.x * (T_ * V_));

  float4 acc; acc.x = 0.f; acc.y = 0.f; acc.z = 0.f; acc.w = 0.f;
#pragma unroll
  for (int j = 0; j < 4; ++j) {
    float4 v = xp4[j * 256 + tid];          // coalesced b128
    acc.x += v.x; acc.y += v.y; acc.z += v.z; acc.w += v.w;
  }
  red[tid] = acc;
  __syncthreads();

  // thread i accumulated v-quad (i%16); partials live at i%16 + 16*p
  if (tid < 16) {
    float4 s; s.x = 0.f; s.y = 0.f; s.z = 0.f; s.w = 0.f;
#pragma unroll
    for (int p = 0; p < 16; ++p) {
      float4 v = red[tid + 16 * p];
      s.x += v.x; s.y += v.y; s.z += v.z; s.w += v.w;
    }
    const float inv = 1.0f / (float)T_;
    s.x *= inv; s.y *= inv; s.z *= inv; s.w *= inv;
    ((float4*)xm)[(size_t)blockIdx.x * 16 + tid] = s;
  }
}

// ---------------------------------------------------------------------------
// Kernel 2: one block per n (256 threads = 8 waves).
//   q(32x64)   = Wq(32x256) @ xmean_n(256x64) + bq      [WMMA f32 16x16x4]
//   q         /= max(||q||_col, eps) ; fold in INTER^-0.5
//   H(64x128)  = q^T @ proto(32x128)                    [WMMA f32 16x16x4]
//   per row: top-10 -> softmax -> scatter, zeros elsewhere
// ---------------------------------------------------------------------------
__global__ __launch_bounds__(256) void hypergraph_kernel(
    const float* __restrict__ xm,     // (N, C, V)
    const float* __restrict__ Wq,     // (INTER, C)
    const float* __restrict__ bq,     // (INTER)
    const float* __restrict__ proto,  // (INTER, M)
    float* __restrict__ out)          // (N, V, M)
{
  __shared__ float q[INTER_ * 65];    // pitch 65 (bank-conflict pad)
  __shared__ float H[V_ * 129];       // pitch 129

  const int n      = blockIdx.x;
  const int tid    = threadIdx.x;
  const int w      = tid >> 5;        // wave id 0..7
  const int lane   = tid & 31;
  const int laneM  = lane & 15;
  const int half   = lane >> 4;       // 0: lanes 0-15, 1: lanes 16-31
  const int koff   = half * 2;        // A/B K-offset per ISA 16x4 f32 layout
  const int rowoff = half * 8;        // C/D row offset per ISA 16x16 layout

  const float* xmn = xm + (size_t)n * (C_ * V_);

  // ---- GEMM1: q = Wq @ xmean_n + bq ----
  {
    const int o0 = (w >> 2) * 16;     // 0 / 16
    const int v0 = (w & 3) * 16;      // 0..48
    v8f c = {0.f, 0.f, 0.f, 0.f, 0.f, 0.f, 0.f, 0.f};
    const float* wrow = Wq + (size_t)(o0 + laneM) * C_;
#pragma unroll 4
    for (int k0 = 0; k0 < C_; k0 += 4) {
      v2f a, b;
      a.x = wrow[k0 + koff];
      a.y = wrow[k0 + koff + 1];
      b.x = xmn[(k0 + koff)     * V_ + v0 + laneM];
      b.y = xmn[(k0 + koff + 1) * V_ + v0 + laneM];
      c = __builtin_amdgcn_wmma_f32_16x16x4_f32(false, a, false, b,
                                                (short)0, c, false, false);
    }
#pragma unroll
    for (int r = 0; r < 8; ++r) {
      const int row = o0 + r + rowoff;
      q[row * 65 + v0 + laneM] = c[r] + bq[row];
    }
  }
  __syncthreads();

  // ---- column-wise L2 normalize over channels; fold in scale = 32^-0.5 ----
  if (tid < V_) {
    float ss = 0.f;
#pragma unroll
    for (int o = 0; o < INTER_; ++o) { float t = q[o * 65 + tid]; ss += t * t; }
    float nrm = fmaxf(sqrtf(ss), 1e-12f);
    float inv = 0.17677669529663687f / nrm;   // INTER^-0.5 / norm
#pragma unroll
    for (int o = 0; o < INTER_; ++o) q[o * 65 + tid] *= inv;
  }
  __syncthreads();

  // ---- GEMM2: H = q^T @ proto ----
#pragma unroll
  for (int j = 0; j < 4; ++j) {
    const int t  = w * 4 + j;          // 32 tiles, disjoint across waves
    const int v0 = (t >> 3) * 16;
    const int m0 = (t & 7) * 16;
    v8f c = {0.f, 0.f, 0.f, 0.f, 0.f, 0.f, 0.f, 0.f};
#pragma unroll
    for (int k0 = 0; k0 < INTER_; k0 += 4) {
      v2f a, b;
      a.x = q[(k0 + koff)     * 65 + v0 + laneM];   // A[v][k] = q[k][v]
      a.y = q[(k0 + koff + 1) * 65 + v0 + laneM];
      b.x = proto[(k0 + koff)     * M_ + m0 + laneM];
      b.y = proto[(k0 + koff + 1) * M_ + m0 + laneM];
      c = __builtin_amdgcn_wmma_f32_16x16x4_f32(false, a, false, b,
                                                (short)0, c, false, false);
    }
#pragma unroll
    for (int r = 0; r < 8; ++r)
      H[(v0 + r + rowoff) * 129 + m0 + laneM] = c[r];
  }
  __syncthreads();

  // ---- top-10 + softmax + scatter per row (one thread per v) ----
  if (tid < V_) {
    float tv[K_TOP]; int ti[K_TOP];
#pragma unroll
    for (int j = 0; j < K_TOP; ++j) { tv[j] = -1e30f; ti[j] = 0; }
    float* hrow = &H[tid * 129];
    for (int m = 0; m < M_; ++m) {
      const float v = hrow[m];
      if (v > tv[K_TOP - 1]) {           // strict > : ties keep earlier index
        int pos = K_TOP - 1;
#pragma unroll
        for (int j = K_TOP - 2; j >= 0; --j) if (v > tv[j]) pos = j;
#pragma unroll
        for (int j = K_TOP - 1; j > 0; --j)
          if (j > pos) { tv[j] = tv[j - 1]; ti[j] = ti[j - 1]; }
#pragma unroll
        for (int j = 0; j < K_TOP; ++j)
          if (j == pos) { tv[j] = v; ti[j] = m; }
      }
    }
    const float mx = tv[0];
    float e[K_TOP]; float s = 0.f;
#pragma unroll
    for (int j = 0; j < K_TOP; ++j) { e[j] = expf(tv[j] - mx); s += e[j]; }
    const float is = 1.f / s;
    for (int m = 0; m < M_; ++m) hrow[m] = 0.f;
#pragma unroll
    for (int j = 0; j < K_TOP; ++j) hrow[ti[j]] = e[j] * is;
  }
  __syncthreads();

  // ---- coalesced store of the finished (V, M) slab ----
  float* on = out + (size_t)n * (V_ * M_);
  for (int i = tid; i < V_ * M_; i += 256)
    on[i] = H[(i >> 7) * 129 + (i & 127)];
}

// ---------------------------------------------------------------------------
extern "C" void kernel_launch(void* const* d_in, const int* in_sizes, int n_in,
                              void* d_out, int out_size, void* d_ws, size_t ws_size,
                              hipStream_t stream) {
  const float* x     = (const float*)d_in[0];   // (N,C,T,V)
  const float* Wq    = (const float*)d_in[1];   // (INTER,C)
  const float* bq    = (const float*)d_in[2];   // (INTER)
  const float* proto = (const float*)d_in[3];   // (INTER,M)
  float* out = (float*)d_out;                   // (N,V,M)
  float* xm  = (float*)d_ws;                    // N*C*V f32 = 4 MB scratch

  mean_t_kernel<<<N_ * C_, 256, 0, stream>>>(x, xm);
  hypergraph_kernel<<<N_, 256, 0, stream>>>(xm, Wq, bq, proto, out);
}